// RIMModule_76690936037487
// MI455X (gfx1250) — compile-verified
//
#include <hip/hip_runtime.h>
#include <math.h>

typedef __attribute__((ext_vector_type(2))) float v2f;
typedef __attribute__((ext_vector_type(4))) float v4f;
typedef __attribute__((ext_vector_type(8))) float v8f;
typedef __attribute__((ext_vector_type(4))) unsigned int v4u;
typedef __attribute__((ext_vector_type(8))) int v8i;
typedef __attribute__((ext_vector_type(4))) int v4i;

#define KR 8      // RIMs
#define HD 256    // hidden
#define AD 128    // attention dim
#define ID 128    // input dim
#define NACT 3
#define BB 32
#define SS 2048

#define SC 128                 // s-rows per workgroup chunk
#define ST 64                  // s-rows staged in LDS per sub-tile
#define CHUNKS (SS / SC)       // 16
#define SUBT (SC / ST)         // 2

// padded LDS strides (in floats) for bank-conflict-free A/B operand fetches
#define KQ_STRIDE 132
#define XT_STRIDE 132
#define SIM_STRIDE 68

// ---------------------------------------------------------------------------
// Kernel 0: zero the T accumulator (B x 16 x 128 f32)
// ---------------------------------------------------------------------------
__global__ __launch_bounds__(256) void rim_zero_f32(float* __restrict__ p, int n) {
    int i = blockIdx.x * 256 + threadIdx.x;
    if (i < n) p[i] = 0.0f;
}

// ---------------------------------------------------------------------------
// Kernel 1: precompute q -> kq (padded to 16 rows) and hh = rim_hidden@W_hh+b
// ---------------------------------------------------------------------------
__global__ __launch_bounds__(256) void rim_precompute(
    const float* __restrict__ rim_hidden,  // (K,H)
    const float* __restrict__ W_q,         // (K,H,A)
    const float* __restrict__ W_k,         // (K,I,A)
    const float* __restrict__ W_hh,        // (H,H)
    const float* __restrict__ bvec,        // (H,)
    float* __restrict__ kq_out,            // (16,I)  rows 8..15 zero
    float* __restrict__ hh_out)            // (K,H)
{
    __shared__ float q_s[KR][AD];
    const int t = threadIdx.x;

    for (int o = t; o < KR * AD; o += 256) {
        int k = o / AD, a = o % AD;
        const float* rh = rim_hidden + k * HD;
        const float* wq = W_q + (size_t)k * HD * AD + a;
        float acc = 0.0f;
        for (int h = 0; h < HD; ++h) acc += rh[h] * wq[(size_t)h * AD];
        q_s[k][a] = acc;
    }
    __syncthreads();

    for (int o = t; o < 16 * ID; o += 256) {
        int k = o / ID, i = o % ID;
        float acc = 0.0f;
        if (k < KR) {
            const float* wk = W_k + ((size_t)k * ID + i) * AD;
            for (int a = 0; a < AD; ++a) acc += wk[a] * q_s[k][a];
        }
        kq_out[o] = acc;
    }

    for (int o = t; o < KR * HD; o += 256) {
        int k = o / HD, h = o % HD;
        const float* rh = rim_hidden + k * HD;
        float acc = bvec[h];
        for (int hp = 0; hp < HD; ++hp) acc += rh[hp] * W_hh[(size_t)hp * HD + h];
        hh_out[o] = acc;
    }
}

// ---------------------------------------------------------------------------
// TDM: issue a 64x128-f32 2D tile load global->LDS with a hardware-padded
// LDS row stride of 132 floats (pad_interval code 6 = 128 DWORDs between
// pads, pad_amount code 3 = 4 DWORDs).  D# layout per CDNA5 ISA ch.8.
// This toolchain exposes the 6-arg builtin:
//   (uint32x4 g0, int32x8 g1, int32x4 g2, int32x4 g3, int32x8 gx, i32 cpol)
// ---------------------------------------------------------------------------
#if __has_builtin(__builtin_amdgcn_tensor_load_to_lds)
#define HAS_TDM 1
__device__ __forceinline__ void tdm_load_tile(const float* gsrc, unsigned lds_addr) {
    unsigned long long ga = (unsigned long long)(uintptr_t)gsrc;
    v4u g0;
    g0[0] = 1u;                                            // count=1, user descriptor
    g0[1] = lds_addr;                                      // lds_addr[31:0]
    g0[2] = (unsigned)ga;                                  // global_addr[31:0]
    g0[3] = ((unsigned)(ga >> 32) & 0x01FFFFFFu)           // global_addr[56:32]
            | (2u << 30);                                  // type=2 ("image")
    v8i g1;
    g1[0] = (int)((2u << 16)      // data_size = 4 bytes
                | (1u << 20)      // pad_enable
                | (6u << 22)      // pad_interval: 128 DWORDs
                | (3u << 25));    // pad_amount: 4 DWORDs
    g1[1] = (int)(((unsigned)ID) << 16);   // tensor_dim0[15:0]=128 in [31:16]
    g1[2] = (int)(((unsigned)ST) << 16);   // tensor_dim0 hi=0 | tensor_dim1[15:0]=64
    g1[3] = (int)(((unsigned)ID) << 16);   // tensor_dim1 hi=0 | tile_dim0=128
    g1[4] = ST;                             // tile_dim1=64 | tile_dim2=0
    g1[5] = ID;                             // tensor_dim0_stride[31:0] = 128
    g1[6] = 0;                              // stride hi | tensor_dim1_stride lo
    g1[7] = 0;
    v4i gz = {0, 0, 0, 0};                  // groups 2/3 unused (2D tile)
    v8i gx = {0, 0, 0, 0, 0, 0, 0, 0};      // extension group: disabled
    __builtin_amdgcn_tensor_load_to_lds(g0, g1, gz, gz, gx, 0);
}
#else
#define HAS_TDM 0
#endif

// ---------------------------------------------------------------------------
// One staged sub-tile of the heavy pass:
//   Sim(16 x ST) = KqPad(16 x ID) * Xtile^T          (WMMA f32 16x16x4)
//   T(16 x 128) += Sim(16 x ST) * Xtile(ST x 128)    (WMMA f32 16x16x4)
// Caller guarantees xtile is ready in LDS; internal barriers protect sim_s.
// ---------------------------------------------------------------------------
__device__ __forceinline__ void rim_subtile(
    const float* __restrict__ xtile,   // LDS, ST x XT_STRIDE
    const float* __restrict__ kq_s,    // LDS, 16 x KQ_STRIDE
    float* __restrict__ sim_s,         // LDS, 16 x SIM_STRIDE
    int wave, int m16, int kh,
    v8f& accT0, v8f& accT1)
{
    // ---- stage 1: Sim tile for this wave: M = rim k (16), N = s in [16w,16w+16)
    {
        v8f c = {0.f, 0.f, 0.f, 0.f, 0.f, 0.f, 0.f, 0.f};
        const int n = (wave << 4) + m16;   // local s column
        #pragma unroll 8
        for (int i = 0; i < ID; i += 4) {
            v2f a  = *(const v2f*)&kq_s[m16 * KQ_STRIDE + i + 2 * kh];
            v2f bb = *(const v2f*)&xtile[n * XT_STRIDE + i + 2 * kh];
            c = __builtin_amdgcn_wmma_f32_16x16x4_f32(false, a, false, bb,
                                                      (short)0, c, false, false);
        }
        const int M0 = kh << 3;
        #pragma unroll
        for (int r = 0; r < 8; ++r)
            sim_s[(M0 + r) * SIM_STRIDE + n] = c[r];
    }
    __syncthreads();

    // ---- stage 2: T += Sim * Xtile ; wave owns N tiles a0 and a0+16
    {
        const int a0 = wave << 5;
        const int a1 = a0 + 16;
        #pragma unroll 4
        for (int sp = 0; sp < ST; sp += 4) {
            const int r0 = sp + 2 * kh;
            v2f a = *(const v2f*)&sim_s[m16 * SIM_STRIDE + r0];
            v2f b0, b1;
            b0[0] = xtile[(r0)     * XT_STRIDE + a0 + m16];
            b0[1] = xtile[(r0 + 1) * XT_STRIDE + a0 + m16];
            b1[0] = xtile[(r0)     * XT_STRIDE + a1 + m16];
            b1[1] = xtile[(r0 + 1) * XT_STRIDE + a1 + m16];
            accT0 = __builtin_amdgcn_wmma_f32_16x16x4_f32(false, a, false, b0,
                                                          (short)0, accT0, false, false);
            accT1 = __builtin_amdgcn_wmma_f32_16x16x4_f32(false, a, false, b1,
                                                          (short)0, accT1, false, false);
        }
    }
    __syncthreads();   // protects sim_s / xtile reuse
}

// ---------------------------------------------------------------------------
// Kernel 2: the heavy pass.  x is streamed through double-buffered LDS tiles,
// staged by the Tensor Data Mover (TENSORcnt) when available.
// ---------------------------------------------------------------------------
__global__ __launch_bounds__(128) void rim_gemm(
    const float* __restrict__ x,   // (B,S,I)
    const float* __restrict__ kq,  // (16,I)
    float* __restrict__ T)         // (B,16,I) accumulator
{
    __shared__ float kq_s[16 * KQ_STRIDE];
    __shared__ float xt[2][ST * XT_STRIDE];
    __shared__ float sim_s[16 * SIM_STRIDE];

    const int tid  = threadIdx.x;
    const int wave = tid >> 5;       // 0..3
    const int lane = tid & 31;
    const int m16  = lane & 15;
    const int kh   = lane >> 4;      // K-half select (0/1)
    const int b     = blockIdx.x / CHUNKS;
    const int chunk = blockIdx.x % CHUNKS;

    // stage Kq (16 x 128) into LDS with padded stride
    for (int o = tid; o < 16 * ID; o += 128) {
        int r = o >> 7, c = o & 127;
        kq_s[r * KQ_STRIDE + c] = kq[o];
    }

    const float* xg = x + ((size_t)b * SS + (size_t)chunk * SC) * ID;

    v8f accT0 = {0.f, 0.f, 0.f, 0.f, 0.f, 0.f, 0.f, 0.f};
    v8f accT1 = accT0;

#if HAS_TDM
    // Issue both sub-tile DMAs up front (one TDM op per tile, wave 0 only;
    // EXEC is ignored by tensor ops, TENSORcnt completes in order).
    if (wave == 0) {
        tdm_load_tile(xg,               (unsigned)(uintptr_t)(void*)&xt[0][0]);
        tdm_load_tile(xg + ST * ID,     (unsigned)(uintptr_t)(void*)&xt[1][0]);
        __builtin_amdgcn_s_wait_tensorcnt((short)1);   // tile 0 landed
    }
    __syncthreads();
    rim_subtile(xt[0], kq_s, sim_s, wave, m16, kh, accT0, accT1);
    if (wave == 0) {
        __builtin_amdgcn_s_wait_tensorcnt((short)0);   // tile 1 landed
    }
    __syncthreads();
    rim_subtile(xt[1], kq_s, sim_s, wave, m16, kh, accT0, accT1);
#else
    for (int st = 0; st < SUBT; ++st) {
        const float* src = xg + (size_t)st * ST * ID;
        float* dst = xt[st & 1];
        {
            int row = tid >> 5;          // 0..3
            int col = (tid & 31) << 2;   // 0..124
            #pragma unroll 4
            for (int it = 0; it < ST / 4; ++it) {
                v4f v = *(const v4f*)(src + (size_t)row * ID + col);
                *(v4f*)&dst[row * XT_STRIDE + col] = v;
                row += 4;
            }
        }
        __syncthreads();
        if (st + 1 < SUBT) {
            __builtin_prefetch((const char*)(xg + (size_t)(st + 1) * ST * ID) + tid * 256, 0, 1);
        }
        rim_subtile(dst, kq_s, sim_s, wave, m16, kh, accT0, accT1);
    }
#endif

    // combine partial T tiles across s-chunks with f32 atomics
    float* Tb = T + (size_t)b * 16 * ID;
    const int M0 = kh << 3;
    const int n0 = (wave << 5) + m16;
    #pragma unroll
    for (int r = 0; r < 8; ++r) {
        atomicAdd(&Tb[(M0 + r) * ID + n0],      accT0[r]);
        atomicAdd(&Tb[(M0 + r) * ID + n0 + 16], accT1[r]);
    }
}

// ---------------------------------------------------------------------------
// Kernel 3: finalize.  attended = T[b,k,:] @ W_v[k]; cand = tanh(att@W_ih + hh);
// mask is the constant k<NACT (top-k over the all-zero padded sim position).
// ---------------------------------------------------------------------------
__global__ __launch_bounds__(128) void rim_finalize(
    const float* __restrict__ T,          // (B,16,I)
    const float* __restrict__ W_v,        // (K,I,A)
    const float* __restrict__ W_ih,       // (A,H)
    const float* __restrict__ hh,         // (K,H)
    const float* __restrict__ rim_hidden, // (K,H)
    float* __restrict__ out)              // (B,K,H)
{
    __shared__ float att_s[AD];
    const int bk = blockIdx.x;
    const int b  = bk / KR;
    const int k  = bk % KR;
    const float* t_row = T + ((size_t)b * 16 + k) * ID;

    {
        const int a = threadIdx.x;
        const float* wv = W_v + (size_t)k * ID * AD + a;
        float acc = 0.0f;
        for (int i = 0; i < ID; ++i) acc += t_row[i] * wv[(size_t)i * AD];
        att_s[a] = acc;
    }
    __syncthreads();

    const bool active = (k < NACT);
    for (int h = threadIdx.x; h < HD; h += 128) {
        float c = hh[k * HD + h];
        for (int a = 0; a < AD; ++a) c += att_s[a] * W_ih[(size_t)a * HD + h];
        float cv = tanhf(c);
        float rh = rim_hidden[k * HD + h];
        out[((size_t)b * KR + k) * HD + h] = active ? cv : rh;
    }
}

// ---------------------------------------------------------------------------
extern "C" void kernel_launch(void* const* d_in, const int* in_sizes, int n_in,
                              void* d_out, int out_size, void* d_ws, size_t ws_size,
                              hipStream_t stream) {
    (void)in_sizes; (void)n_in; (void)out_size; (void)ws_size;

    const float* x    = (const float*)d_in[0];
    const float* rim  = (const float*)d_in[1];
    const float* W_q  = (const float*)d_in[2];
    const float* W_k  = (const float*)d_in[3];
    const float* W_v  = (const float*)d_in[4];
    const float* W_ih = (const float*)d_in[5];
    const float* W_hh = (const float*)d_in[6];
    const float* bvec = (const float*)d_in[7];

    // workspace layout (floats): kq(16*128) | hh(8*256) | T(32*16*128)
    float* wsf = (float*)d_ws;
    float* kq  = wsf;
    float* hh  = wsf + 16 * ID;
    float* T   = hh + KR * HD;

    const int tElems = BB * 16 * ID;
    rim_zero_f32<<<(tElems + 255) / 256, 256, 0, stream>>>(T, tElems);
    rim_precompute<<<1, 256, 0, stream>>>(rim, W_q, W_k, W_hh, bvec, kq, hh);
    rim_gemm<<<BB * CHUNKS, 128, 0, stream>>>(x, kq, T);
    rim_finalize<<<BB * KR, 128, 0, stream>>>(T, W_v, W_ih, hh, rim, (float*)d_out);
}